// FaceTokenModel_4664334483526
// MI455X (gfx1250) — compile-verified
//
#include <hip/hip_runtime.h>
#include <math.h>

typedef float v2f __attribute__((ext_vector_type(2)));
typedef float v8f __attribute__((ext_vector_type(8)));

// ---------------- compile-time PGA(3,0,1) tables (metric: e0^2 = 0) -------------
struct GaTables {
  float gpc[16][16]; int gpo[16][16];   // geometric product coef / output blade
  float jnc[16][16]; int jno[16][16];   // join (dual-wedge-dual) coef / output blade
  int   mask[16];    int grade[16];     // blade index -> generator bitmask / grade
  int   idx_of_mask[16];                // bitmask -> blade index
};

constexpr int popc4(unsigned x){ int c=0; for(int i=0;i<4;++i) c += (x>>i)&1; return c; }
constexpr int rsign(unsigned a, unsigned b){
  int s = 0;
  for (int j=0;j<4;++j) if ((b>>j)&1) s += popc4(a >> (j+1));
  return (s&1) ? -1 : 1;
}
constexpr GaTables make_tables(){
  GaTables t{};
  const unsigned M[16] = {0,1,2,4,8,3,5,9,6,10,12,7,11,13,14,15}; // BLADES order
  int IDX[16] = {};
  for (int i=0;i<16;++i) IDX[(int)M[i]] = i;
  for (int i=0;i<16;++i){ t.mask[i]=(int)M[i]; t.grade[i]=popc4(M[i]); t.idx_of_mask[(int)M[i]] = i; }
  for (int i=0;i<16;++i) for (int j=0;j<16;++j){
    unsigned a=M[i], b=M[j];
    if (a & b & 1u){ t.gpc[i][j]=0.f; t.gpo[i][j]=0; }               // e0*e0 = 0
    else           { t.gpc[i][j]=(float)rsign(a,b); t.gpo[i][j]=IDX[(int)(a^b)]; }
    unsigned da = 15u ^ a, db = 15u ^ b;
    if (da & db){ t.jnc[i][j]=0.f; t.jno[i][j]=0; }
    else {
      int si = rsign(a, da), sj = rsign(b, db), sw = rsign(da, db);
      unsigned w = da ^ db, km = 15u ^ w;
      int sk = rsign(km, w);
      t.jnc[i][j] = (float)(si*sj*sw*sk);
      t.jno[i][j] = IDX[(int)km];
    }
  }
  return t;
}
constexpr GaTables GA = make_tables();

// ---------------- tiling ----------------
#define TOK_PER_WAVE 16
#define WAVES_PER_BLK 2
#define THREADS (32*WAVES_PER_BLK)
#define STATE_F (16*16*16)   // [ch up to 16][16 comps][16 tokens] floats

// equi_linear on a 16-token tile via V_WMMA_F32_16X16X4_F32.
//   A (M=token x K=in-ch)  gathered per component from LDS state
//   B (K=in-ch x N=out-ch) : only 9 distinct fragments (one per weight slice) --
//                            preloaded into registers once, reused by all comps
//   D (M=token x N=out-ch) stored unconditionally (cols >= OC are never read)
// state layout: s[(ch*16 + comp)*16 + token]
template<int IC, int OC>
__device__ __forceinline__ void equi_linear_t(const float* __restrict__ sin,
    float* __restrict__ sout, const float* __restrict__ W,
    int l15, int half)
{
  constexpr int CH = IC/4;                      // K chunks of 4
  const int n = l15 & (OC-1);                   // OC is a power of two: clamp in-bounds
  v2f Bf[9][CH];
  #pragma unroll
  for (int b = 0; b < 9; ++b)
    #pragma unroll
    for (int c = 0; c < CH; ++c){
      const int K0 = c*4 + 2*half;              // B: N=l15, VGPR v holds K=K0+v
      Bf[b][c][0] = W[(n*IC + K0+0)*9 + b];
      Bf[b][c][1] = W[(n*IC + K0+1)*9 + b];
    }
  #pragma unroll
  for (int k = 0; k < 16; ++k){                 // fully unrolled: g/km fold
    v8f acc = {0.f,0.f,0.f,0.f,0.f,0.f,0.f,0.f};
    const int g  = GA.grade[k];
    const int km = GA.mask[k];
    #pragma unroll
    for (int c = 0; c < CH; ++c){
      const int K0 = c*4 + 2*half;              // A: M=l15, VGPR v holds K=K0+v
      v2f A;
      A[0] = sin[((K0+0)*16 + k)*16 + l15];
      A[1] = sin[((K0+1)*16 + k)*16 + l15];
      acc = __builtin_amdgcn_wmma_f32_16x16x4_f32(
          false, A, false, Bf[g][c], (short)0, acc, false, false);
    }
    if (km & 1){                                // compile-time after unroll
      const int src = GA.idx_of_mask[km & ~1];  // e0-extension source blade
      #pragma unroll
      for (int c = 0; c < CH; ++c){
        const int K0 = c*4 + 2*half;
        v2f A;
        A[0] = sin[((K0+0)*16 + src)*16 + l15];
        A[1] = sin[((K0+1)*16 + src)*16 + l15];
        acc = __builtin_amdgcn_wmma_f32_16x16x4_f32(
            false, A, false, Bf[4+g][c], (short)0, acc, false, false);
      }
    }
    #pragma unroll
    for (int v = 0; v < 8; ++v)                 // D: M = v + 8*half, N = l15
      sout[(l15*16 + k)*16 + (v + half*8)] = acc[v];
  }
}

__device__ __forceinline__ void rmsnorm_tile(float* s, int ch, int l15, int half){
  float ms = 0.f;
  for (int c = 0; c < ch; ++c)
    #pragma unroll
    for (int k = 0; k < 16; ++k){ float v = s[(c*16+k)*16 + l15]; ms += v*v; }
  const float inv = rsqrtf(ms/(float)ch + 1e-9f);
  const int c0 = half*(ch>>1), c1 = c0 + (ch>>1);
  for (int c = c0; c < c1; ++c)
    #pragma unroll
    for (int k = 0; k < 16; ++k) s[(c*16+k)*16 + l15] *= inv;
}

__device__ __forceinline__ float gelu_tanh(float x){
  const float x3 = x*x*x;
  return 0.5f*x*(1.f + tanhf(0.7978845608028654f*(x + 0.044715f*x3)));
}

__device__ __forceinline__ void gate_tile(float* s, int ch, int l15, int half){
  const int c0 = half*(ch>>1), c1 = c0 + (ch>>1);
  for (int c = c0; c < c1; ++c){
    const float g = gelu_tanh(s[(c*16+0)*16 + l15]);
    #pragma unroll
    for (int k = 0; k < 16; ++k) s[(c*16+k)*16 + l15] *= g;
  }
}

// 16-ch input -> 8-ch output: 4 GP pairs + 4 join pairs (scaled by r15)
__device__ __forceinline__ void gp_join_tile(const float* sin, float* sout,
                                             float r15, int l15, int half)
{
  for (int c = half*2; c < half*2 + 2; ++c){
    float a[16], b[16], o[16];
    #pragma unroll
    for (int k=0;k<16;++k){ a[k]=sin[((0+c)*16+k)*16+l15]; b[k]=sin[((4+c)*16+k)*16+l15]; o[k]=0.f; }
    #pragma unroll
    for (int i=0;i<16;++i)
      #pragma unroll
      for (int j=0;j<16;++j)
        if (GA.gpc[i][j] != 0.f) o[GA.gpo[i][j]] += GA.gpc[i][j]*a[i]*b[j];
    #pragma unroll
    for (int k=0;k<16;++k) sout[(c*16+k)*16+l15] = o[k];

    #pragma unroll
    for (int k=0;k<16;++k){ a[k]=sin[((8+c)*16+k)*16+l15]; b[k]=sin[((12+c)*16+k)*16+l15]; o[k]=0.f; }
    #pragma unroll
    for (int i=0;i<16;++i)
      #pragma unroll
      for (int j=0;j<16;++j)
        if (GA.jnc[i][j] != 0.f) o[GA.jno[i][j]] += GA.jnc[i][j]*a[i]*b[j];
    #pragma unroll
    for (int k=0;k<16;++k) sout[((4+c)*16+k)*16+l15] = o[k]*r15;
  }
}

__device__ __forceinline__ void face_layer_t(float* sS, float* sT,
    const float* pb, const float* po, const float* pp,
    float r15, int l15, int half)
{
  rmsnorm_tile(sS, 8, l15, half);                    // in place
  equi_linear_t<8,16>(sS, sT, pb, l15, half);        // 8 -> 16 ch
  gp_join_tile(sT, sS, r15, l15, half);              // 16 -> 8 ch
  equi_linear_t<8,8>(sS, sT, po, l15, half);         // 8 -> 8
  gate_tile(sT, 8, l15, half);                       // scalar-gated gelu
  equi_linear_t<8,8>(sT, sS, pp, l15, half);         // 8 -> 8 ; result back in sS
}

__global__ __launch_bounds__(THREADS)
void facetoken_fused_kernel(
    const float* __restrict__ f,
    const float* __restrict__ w_proj_i,
    const float* __restrict__ enc_pb, const float* __restrict__ enc_po, const float* __restrict__ enc_pp,
    const float* __restrict__ w_proj_x, const float* __restrict__ w_proj_z,
    const float* __restrict__ dec_pb, const float* __restrict__ dec_po, const float* __restrict__ dec_pp,
    const float* __restrict__ w_proj_o,
    float* __restrict__ out_main, float* __restrict__ out_x, int ntok)
{
  __shared__ float smem[WAVES_PER_BLK*2*STATE_F];  // 64 KB
  const int lane = threadIdx.x & 31;
  const int wave = threadIdx.x >> 5;
  const int l15  = lane & 15;
  const int half = lane >> 4;
  float* sA = &smem[(wave*2+0)*STATE_F];
  float* sB = &smem[(wave*2+1)*STATE_F];

  const long tile0 = ((long)blockIdx.x * WAVES_PER_BLK + wave) * TOK_PER_WAVE;
  if (tile0 >= ntok) return;                       // wave-uniform

  // ---- load f tile: (16 tok x 4 ch x 16 comps), float4-coalesced ----
  #pragma unroll
  for (int i4 = 0; i4 < TOK_PER_WAVE*4*16/4/32; ++i4){
    const int j   = i4*32 + lane;
    const int idx = j*4;
    const int t = idx >> 6, c = (idx >> 4) & 3, k0 = idx & 15;
    const float4 v = ((const float4*)f)[tile0*16 + j];
    sA[(c*16 + k0+0)*16 + t] = v.x;
    sA[(c*16 + k0+1)*16 + t] = v.y;
    sA[(c*16 + k0+2)*16 + t] = v.z;
    sA[(c*16 + k0+3)*16 + t] = v.w;
  }

  // r = mean over input channels; only pseudoscalar component (blade 15) is used
  float r15 = 0.f;
  for (int c = 0; c < 4; ++c) r15 += sA[(c*16+15)*16 + l15];
  r15 *= 0.25f;

  // ---- encoder ----
  equi_linear_t<4,8>(sA, sB, w_proj_i, l15, half);         // state in sB (8 ch)
  for (int l = 0; l < 3; ++l)
    face_layer_t(sB, sA, enc_pb + l*16*8*9, enc_po + l*8*8*9, enc_pp + l*8*8*9,
                 r15, l15, half);

  // ---- bottleneck: x (second output) ----
  equi_linear_t<8,16>(sB, sA, w_proj_x, l15, half);        // x in sA (16 ch)
  #pragma unroll
  for (int i4 = 0; i4 < TOK_PER_WAVE*16*16/4/32; ++i4){
    const int j   = i4*32 + lane;
    const int idx = j*4;
    const int t = idx >> 8, c = (idx >> 4) & 15, k0 = idx & 15;
    float4 v;
    v.x = sA[(c*16 + k0+0)*16 + t];
    v.y = sA[(c*16 + k0+1)*16 + t];
    v.z = sA[(c*16 + k0+2)*16 + t];
    v.w = sA[(c*16 + k0+3)*16 + t];
    ((float4*)out_x)[tile0*64 + j] = v;
  }
  equi_linear_t<16,8>(sA, sB, w_proj_z, l15, half);        // z in sB (8 ch)

  float r2 = 0.f;
  for (int c = 0; c < 8; ++c) r2 += sB[(c*16+15)*16 + l15];
  r2 *= 0.125f;

  // ---- decoder ----
  for (int l = 0; l < 3; ++l)
    face_layer_t(sB, sA, dec_pb + l*16*8*9, dec_po + l*8*8*9, dec_pp + l*8*8*9,
                 r2, l15, half);

  // ---- output projection ----
  equi_linear_t<8,4>(sB, sA, w_proj_o, l15, half);         // out in sA (4 ch)
  #pragma unroll
  for (int i4 = 0; i4 < TOK_PER_WAVE*4*16/4/32; ++i4){
    const int j   = i4*32 + lane;
    const int idx = j*4;
    const int t = idx >> 6, c = (idx >> 4) & 3, k0 = idx & 15;
    float4 v;
    v.x = sA[(c*16 + k0+0)*16 + t];
    v.y = sA[(c*16 + k0+1)*16 + t];
    v.z = sA[(c*16 + k0+2)*16 + t];
    v.w = sA[(c*16 + k0+3)*16 + t];
    ((float4*)out_main)[tile0*16 + j] = v;
  }
}

extern "C" void kernel_launch(void* const* d_in, const int* in_sizes, int n_in,
                              void* d_out, int out_size, void* d_ws, size_t ws_size,
                              hipStream_t stream)
{
  const float* f        = (const float*)d_in[0];
  const float* w_proj_i = (const float*)d_in[1];
  const float* enc_pb   = (const float*)d_in[2];
  const float* enc_po   = (const float*)d_in[3];
  const float* enc_pp   = (const float*)d_in[4];
  const float* w_proj_x = (const float*)d_in[5];
  const float* w_proj_z = (const float*)d_in[6];
  const float* dec_pb   = (const float*)d_in[7];
  const float* dec_po   = (const float*)d_in[8];
  const float* dec_pp   = (const float*)d_in[9];
  const float* w_proj_o = (const float*)d_in[10];

  const int ntok = in_sizes[0] / (4*16);            // B = 131072
  float* out_main = (float*)d_out;                  // (B, 4, 16)
  float* out_x    = out_main + (long)ntok*4*16;     // (B, 16, 16)

  const int tiles  = (ntok + TOK_PER_WAVE - 1) / TOK_PER_WAVE;
  const int blocks = (tiles + WAVES_PER_BLK - 1) / WAVES_PER_BLK;

  facetoken_fused_kernel<<<blocks, THREADS, 0, stream>>>(
      f, w_proj_i, enc_pb, enc_po, enc_pp, w_proj_x, w_proj_z,
      dec_pb, dec_po, dec_pp, w_proj_o, out_main, out_x, ntok);
}